// ISDALoss_53102975647776
// MI455X (gfx1250) — compile-verified
//
#include <hip/hip_runtime.h>

// ISDA loss for MI455X (gfx1250, wave32, WMMA).
// Outputs: d_out[0] = loss, d_out[1..] = y[N=1024, C=1000] row-major.
//
// Heavy math: per-class quadratic form S[l,c] = (w_c - w_l) M_l (w_c - w_l)^T
// done as a 1000x128x128 f32 GEMM per class with v_wmma_f32_16x16x4_f32,
// with M_l (64KB) resident in LDS, staged via global_load_async_to_lds_b128.

#define NS   1024   // samples
#define AF   128    // feature dim
#define CC   1000   // classes
#define CPAD 1024   // padded class stride for S

typedef float v2f __attribute__((ext_vector_type(2)));
typedef float v8f __attribute__((ext_vector_type(8)));

#ifndef __has_builtin
#define __has_builtin(x) 0
#endif

#if __has_builtin(__builtin_amdgcn_global_load_async_to_lds_b128) && \
    __has_builtin(__builtin_amdgcn_s_wait_asynccnt)
#define HAVE_ASYNC_LDS 1
// b128 payload type + explicit address-space pointers, per hipcc's own
// diagnostic: param 0 is "int __vector_size__(16) * __device__" (AS1).
typedef int v4i_b128 __attribute__((vector_size(16)));
typedef __attribute__((address_space(1))) v4i_b128* gbl_b128_p;
typedef __attribute__((address_space(3))) v4i_b128* lds_b128_p;
#else
#define HAVE_ASYNC_LDS 0
#endif

__device__ __forceinline__ v8f wmma_f32(v2f a, v2f b, v8f c) {
  // (neg_a, A, neg_b, B, c_mod, C, reuse_a, reuse_b)
  return __builtin_amdgcn_wmma_f32_16x16x4_f32(false, a, false, b, (short)0, c,
                                               false, false);
}

// ---------------------------------------------------------------------------
// y = X @ W^T + bias.  M=1024 (samples) x N=1000 (classes, pad 1008) x K=128.
// 8 waves/block, each wave one 16x16 tile.  grid = (1024/128, 1008/16)=(8,63).
// Out-of-range class columns are CLAMPED (not predicated): a GEMM output
// column depends only on its own B column, so clamped lanes produce finite
// garbage in never-stored columns -- no EXEC-mask branches in the K-loop.
// ---------------------------------------------------------------------------
__global__ __launch_bounds__(256)
void ygemm_kernel(const float* __restrict__ feat, const float* __restrict__ W,
                  const float* __restrict__ bias, float* __restrict__ y) {
  const int lane = threadIdx.x & 31;
  const int wave = threadIdx.x >> 5;   // 0..7
  const int lr = lane & 15, hi = lane >> 4;
  const int m0 = blockIdx.x * 128 + wave * 16;  // sample rows
  const int c0 = blockIdx.y * 16;               // class cols

  v8f acc = {};
  const int arow = m0 + lr;
  const int bcolRaw = c0 + lr;
  const bool bok = bcolRaw < CC;
  const int bcol = bok ? bcolRaw : (CC - 1);    // clamp: garbage cols unstored

  const float* __restrict__ arow_p = feat + arow * AF;
  const float* __restrict__ bcol_p = W + bcol * AF;

  for (int k0 = 0; k0 < AF; k0 += 4) {
    const int ka = k0 + 2 * hi;
    v2f a, b;
    // A 16x4 f32 frag: lanes 0-15 -> K=k0,k0+1 ; lanes 16-31 -> K=k0+2,k0+3
    a.x = arow_p[ka];
    a.y = arow_p[ka + 1];
    // B 4x16 frag: B[k][c] = W[c][k]
    b.x = bcol_p[ka];
    b.y = bcol_p[ka + 1];
    acc = wmma_f32(a, b, acc);
  }

  if (bok) {
    const float bv = bias[bcolRaw];
#pragma unroll
    for (int r = 0; r < 8; ++r) {
      const int row = m0 + r + 8 * hi;  // C/D layout: VGPR r -> M=r (+8 hi)
      y[row * CC + bcolRaw] = acc[r] + bv;
    }
  }
}

// ---------------------------------------------------------------------------
// One block per class l (grid = 1000, 256 threads = 8 waves):
//  0) kick async DMA of CoV_l (64KB) -> Mlds, overlapped with (1)
//  1) scan labels -> n_l, per-feature sums -> ave_l, w_l, d_ave_l
//  2) M_l = (1-w)*CoV_l + w*var_l + w(1-w) dA dA^T, in place in LDS
//  3) S[l, c] = rowdot( (W - w_l) @ M_l , (W - w_l) )  via f32 WMMA
// ---------------------------------------------------------------------------
__global__ __launch_bounds__(256)
void scov_kernel(const float* __restrict__ feat, const int* __restrict__ labels,
                 const float* __restrict__ W, const float* __restrict__ CoV,
                 const float* __restrict__ AveIn,
                 const float* __restrict__ AmountIn, float* __restrict__ S) {
  __shared__ int   lab[NS];            // 4 KB
  __shared__ float Mlds[AF * AF];      // 64 KB
  __shared__ float ave_row[AF];
  __shared__ float dA_row[AF];
  __shared__ float wl_row[AF];
  __shared__ float cent_row[AF];

  const int l = blockIdx.x;
  const int t = threadIdx.x;

#if HAVE_ASYNC_LDS
  // -------- 0) async-stage CoV_l into Mlds; overlaps the stats scan --------
  {
    const char* gsrc = (const char*)(CoV + (size_t)l * AF * AF);
    char* ldst = (char*)Mlds;
#pragma unroll
    for (int j = 0; j < 16; ++j) {
      const int byteoff = (t + 256 * j) * 16;  // 256 thr * 16 * 16B = 64 KB
      __builtin_amdgcn_global_load_async_to_lds_b128(
          (gbl_b128_p)(gsrc + byteoff), (lds_b128_p)(ldst + byteoff), 0, 0);
    }
  }
#endif

  for (int i = t; i < NS; i += 256) lab[i] = labels[i];
  __syncthreads();

  // -------- 1) per-class stats (all threads scan; threads 0..127 own a=t) --
  float suma = 0.f;
  int cnt = 0;
  for (int n = 0; n < NS; ++n) {
    if (lab[n] == l) {
      ++cnt;
      if (t < AF) suma += feat[n * AF + t];
    }
  }
  const float counts = (float)cnt;
  const float amt = (cnt == 0) ? 1.f : counts;       // amount_c
  const float denom = counts + AmountIn[l];
  const float w = (denom == 0.f) ? 0.f : counts / denom;
  const float one_m_w = 1.f - w;
  const float w1w = w * one_m_w;
  const float wscale = w / amt;                      // w * (1/amount_c)

  if (t < AF) {
    const float av = suma / amt;
    ave_row[t] = av;
    dA_row[t] = AveIn[l * AF + t] - av;              // Ave - ave
    wl_row[t] = W[l * AF + t];
  }

#if HAVE_ASYNC_LDS
  __builtin_amdgcn_s_wait_asynccnt(0);   // my async writes to Mlds landed
#endif
  __syncthreads();                       // everyone's writes visible

  // -------- 2) M_l = (1-w)*CoV + w(1-w)*dA dA^T ----------------------------
  for (int j = 0; j < 64; ++j) {
    const int e = t + 256 * j;
    const int a = e >> 7, b = e & 127;
#if HAVE_ASYNC_LDS
    Mlds[e] = one_m_w * Mlds[e] + w1w * dA_row[a] * dA_row[b];
#else
    Mlds[e] = one_m_w * CoV[(size_t)l * AF * AF + e] +
              w1w * dA_row[a] * dA_row[b];
#endif
  }
  __syncthreads();

  // -------- += w * var_temp  (outer products of centered member rows) ------
  for (int n = 0; n < NS; ++n) {
    if (lab[n] == l) {                 // block-uniform branch: barriers legal
      if (t < AF) cent_row[t] = feat[n * AF + t] - ave_row[t];
      __syncthreads();
      for (int j = 0; j < 64; ++j) {
        const int e = t + 256 * j;
        const int a = e >> 7, b = e & 127;
        Mlds[e] += wscale * cent_row[a] * cent_row[b];
      }
      __syncthreads();
    }
  }
  __syncthreads();

  // -------- 3) WMMA: T = D @ M_l fused with S[l,row] = sum_col T*D ---------
  const int lane = t & 31, wave = t >> 5;
  const int lr = lane & 15, hi = lane >> 4;

  for (int pass = 0; pass < 8; ++pass) {
    const int rbase = pass * 128 + wave * 16;  // class-row tile base, 0..1023
    v8f acc[8] = {};

    const int drowRaw = rbase + lr;
    const int drow = drowRaw < CC ? drowRaw : (CC - 1);  // clamp, no branches
    const float* __restrict__ drow_p = W + (size_t)drow * AF;

    for (int k0 = 0; k0 < AF; k0 += 4) {
      const int ka = k0 + 2 * hi;
      v2f a;
      a.x = drow_p[ka]     - wl_row[ka];
      a.y = drow_p[ka + 1] - wl_row[ka + 1];
#pragma unroll
      for (int ct = 0; ct < 8; ++ct) {
        const int col = ct * 16 + lr;
        v2f b;
        b.x = Mlds[ka * AF + col];
        b.y = Mlds[(ka + 1) * AF + col];
        acc[ct] = wmma_f32(a, b, acc[ct]);
      }
    }

#pragma unroll
    for (int r = 0; r < 8; ++r) {
      const int rowRaw = rbase + r + 8 * hi;
      const int row = rowRaw < CC ? rowRaw : (CC - 1);   // clamp
      float partial = 0.f;
#pragma unroll
      for (int ct = 0; ct < 8; ++ct) {
        const int col = ct * 16 + lr;
        const float d = W[(size_t)row * AF + col] - wl_row[col];
        partial += acc[ct][r] * d;
      }
      // reduce the 16 lanes of each half (wave32 shuffles)
      partial += __shfl_xor(partial, 1, 32);
      partial += __shfl_xor(partial, 2, 32);
      partial += __shfl_xor(partial, 4, 32);
      partial += __shfl_xor(partial, 8, 32);
      if (lr == 0 && rowRaw < CC) S[(size_t)l * CPAD + rowRaw] = partial;
    }
  }
}

// ---------------------------------------------------------------------------
// Per-sample log-softmax CE over aug_y = y + 0.5*ratio*S[label]. grid = 1024.
// ---------------------------------------------------------------------------
__global__ __launch_bounds__(256)
void rowloss_kernel(const float* __restrict__ y, const int* __restrict__ labels,
                    const float* __restrict__ S, const float* __restrict__ ratio,
                    float* __restrict__ rowloss) {
  __shared__ float red[256];
  __shared__ float aug_lab_sh;
  const int n = blockIdx.x;
  const int t = threadIdx.x;
  const int l = labels[n];
  const float hr = 0.5f * ratio[0];

  float mx = -3.402823466e38f;
  for (int c = t; c < CC; c += 256) {
    const float aug = y[(size_t)n * CC + c] + hr * S[(size_t)l * CPAD + c];
    mx = fmaxf(mx, aug);
    if (c == l) aug_lab_sh = aug;  // exactly one thread hits this
  }
  red[t] = mx;
  __syncthreads();
  for (int s = 128; s > 0; s >>= 1) {
    if (t < s) red[t] = fmaxf(red[t], red[t + s]);
    __syncthreads();
  }
  const float rmax = red[0];
  __syncthreads();

  float se = 0.f;
  for (int c = t; c < CC; c += 256) {
    const float aug = y[(size_t)n * CC + c] + hr * S[(size_t)l * CPAD + c];
    se += expf(aug - rmax);
  }
  red[t] = se;
  __syncthreads();
  for (int s = 128; s > 0; s >>= 1) {
    if (t < s) red[t] += red[t + s];
    __syncthreads();
  }
  if (t == 0) rowloss[n] = -(aug_lab_sh - rmax - logf(red[0]));
}

__global__ __launch_bounds__(1024)
void loss_reduce_kernel(const float* __restrict__ rowloss,
                        float* __restrict__ out) {
  __shared__ float red[1024];
  const int t = threadIdx.x;
  red[t] = rowloss[t];
  __syncthreads();
  for (int s = 512; s > 0; s >>= 1) {
    if (t < s) red[t] += red[t + s];
    __syncthreads();
  }
  if (t == 0) out[0] = red[0] / (float)NS;
}

// ---------------------------------------------------------------------------
extern "C" void kernel_launch(void* const* d_in, const int* in_sizes, int n_in,
                              void* d_out, int out_size, void* d_ws,
                              size_t ws_size, hipStream_t stream) {
  const float* feat   = (const float*)d_in[0];  // [1024,128]
  const int*   labels = (const int*)d_in[1];    // [1024] (int32 on device)
  const float* W      = (const float*)d_in[2];  // [1000,128]
  const float* bias   = (const float*)d_in[3];  // [1000]
  const float* CoV    = (const float*)d_in[4];  // [1000,128,128]
  const float* AveIn  = (const float*)d_in[5];  // [1000,128]
  const float* Amount = (const float*)d_in[6];  // [1000]
  const float* ratio  = (const float*)d_in[7];  // [1]

  float* loss_out = (float*)d_out;
  float* y_out    = loss_out + 1;               // [1024,1000]

  float* S       = (float*)d_ws;                // CC * CPAD floats
  float* rowloss = S + (size_t)CC * CPAD;       // NS floats
  (void)in_sizes; (void)n_in; (void)out_size; (void)ws_size;

  ygemm_kernel<<<dim3(8, 63), 256, 0, stream>>>(feat, W, bias, y_out);
  scov_kernel<<<dim3(CC), 256, 0, stream>>>(feat, labels, W, CoV, AveIn, Amount, S);
  rowloss_kernel<<<dim3(NS), 256, 0, stream>>>(y_out, labels, S, ratio, rowloss);
  loss_reduce_kernel<<<dim3(1), 1024, 0, stream>>>(rowloss, loss_out);
}